// Attention_28037546508934
// MI455X (gfx1250) — compile-verified
//
#include <hip/hip_runtime.h>
#include <math.h>

// CDNA5 (gfx1250) fp32 WMMA fragments
typedef float v2f __attribute__((ext_vector_type(2)));
typedef float v8f __attribute__((ext_vector_type(8)));

#define G_    1024
#define N_    512
#define IN_   256
#define H_    128
#define OUT_  128
#define XS_STRIDE 132   // 512 rows * 132 floats: bank-spread LDS layout

// Branch-free tanh: 1 - 2/(e^{2x}+1).  Saturates correctly at +/-inf,
// no EXEC divergence; uses hardware v_rcp_f32 (~1 ulp) instead of the
// IEEE div_scale/div_fmas sequence -- plenty for an activation.
__device__ __forceinline__ float fast_tanh(float x) {
    const float t = __expf(2.0f * x);
    return 1.0f - 2.0f * __builtin_amdgcn_rcpf(t + 1.0f);
}

// ---------------------------------------------------------------------------
// Kernel 1: per-group  x = tanh(feat @ hq_w^T + hq_b); softmax-pool over N.
// One workgroup per group, 8 waves; wave w owns h-columns [16w, 16w+16).
// GEMM via v_wmma_f32_16x16x4_f32; x tile resides in LDS (dynamic, ~272KB).
// ---------------------------------------------------------------------------
extern "C" __global__ void __launch_bounds__(256)
k1_group_pool(const float* __restrict__ feat,
              const float* __restrict__ hq_w, const float* __restrict__ hq_b,
              const float* __restrict__ hk_w, const float* __restrict__ hk_b,
              float* __restrict__ stacked)
{
    extern __shared__ float smem[];
    float* xs  = smem;                       // [512][XS_STRIDE]
    float* ss  = smem + N_ * XS_STRIDE;      // [512] scores / exp
    float* red = ss + N_;                    // [256] reduction scratch

    const int tid   = threadIdx.x;
    const int wave  = tid >> 5;              // wave32
    const int lane  = tid & 31;
    const int lhalf = lane >> 4;             // 0: K={k,k+1}, 1: K={k+2,k+3}
    const int lmod  = lane & 15;             // A: row-in-tile / B,C: column
    const int g     = blockIdx.x;

    const float* fg  = feat + (size_t)g * N_ * IN_;
    const int   h0   = wave * 16;
    const int   col  = h0 + lmod;
    const float bias = hq_b[col];
    const float* bcol = hq_w + (size_t)col * IN_;    // B[k][col] = hq_w[col][k]

    // ---- Phase A: GEMM1 + tanh -> xs in LDS --------------------------------
    for (int mt = 0; mt < 32; ++mt) {
        const int m0 = mt * 16;
        const float* arow = fg + (size_t)(m0 + lmod) * IN_ + 2 * lhalf;
        if (mt < 31)
            __builtin_prefetch(arow + 16 * IN_, 0, 0);   // global_prefetch next m-tile

        v8f acc = {0.f, 0.f, 0.f, 0.f, 0.f, 0.f, 0.f, 0.f};
        #pragma unroll 8
        for (int ks = 0; ks < 64; ++ks) {
            const int k = ks * 4;
            v2f a = *(const v2f*)(arow + k);                 // A 16x4 fragment
            v2f b = *(const v2f*)(bcol + k + 2 * lhalf);     // B 4x16 fragment
            acc = __builtin_amdgcn_wmma_f32_16x16x4_f32(
                false, a, false, b, (short)0, acc, false, false);
        }
        #pragma unroll
        for (int v = 0; v < 8; ++v) {
            const int row = m0 + v + 8 * lhalf;              // C/D layout
            xs[row * XS_STRIDE + col] = fast_tanh(acc[v] + bias);
        }
    }
    __syncthreads();

    // ---- Phase B: scores = xs . hk_w + hk_b --------------------------------
    {
        const float kb = hk_b[0];
        for (int r = tid; r < N_; r += 256) {
            float s = kb;
            for (int k = 0; k < H_; ++k)
                s += xs[r * XS_STRIDE + k] * hk_w[k];        // hk_w: scalar loads
            ss[r] = s;
        }
    }
    __syncthreads();

    // ---- softmax over N=512 ------------------------------------------------
    const float v0 = ss[tid];
    const float v1 = ss[tid + 256];
    red[tid] = fmaxf(v0, v1);
    __syncthreads();
    for (int off = 128; off > 0; off >>= 1) {
        if (tid < off) red[tid] = fmaxf(red[tid], red[tid + off]);
        __syncthreads();
    }
    const float mx = red[0];
    __syncthreads();

    const float e0 = __expf(v0 - mx);
    const float e1 = __expf(v1 - mx);
    ss[tid]       = e0;
    ss[tid + 256] = e1;
    red[tid] = e0 + e1;
    __syncthreads();
    for (int off = 128; off > 0; off >>= 1) {
        if (tid < off) red[tid] += red[tid + off];
        __syncthreads();
    }
    const float Z = red[0];
    __syncthreads();

    // ---- weighted pool: stacked[g][h] = sum_n w[n] * x[n][h] ---------------
    {
        const int h    = tid & 127;
        const int half = tid >> 7;
        const int nb   = half * 256;
        float a = 0.f;
        for (int n = 0; n < 256; ++n)
            a += ss[nb + n] * xs[(nb + n) * XS_STRIDE + h];
        red[tid] = a;
        __syncthreads();
        if (tid < 128)
            stacked[(size_t)g * H_ + tid] = (red[tid] + red[tid + 128]) / Z;
    }
}

// ---------------------------------------------------------------------------
// Kernel 2: y = tanh(stacked @ mq_w^T + mq_b) via WMMA; s2 = y . mk_w + mk_b.
// 8 blocks x 256 threads; block b handles groups [128b, 128b+128).
// ---------------------------------------------------------------------------
extern "C" __global__ void __launch_bounds__(256)
k2_group_mlp(const float* __restrict__ stacked,
             const float* __restrict__ mq_w, const float* __restrict__ mq_b,
             const float* __restrict__ mk_w, const float* __restrict__ mk_b,
             float* __restrict__ y, float* __restrict__ s2)
{
    extern __shared__ float smem2[];
    float* ys = smem2;                        // [128][129]

    const int tid   = threadIdx.x;
    const int wave  = tid >> 5;
    const int lane  = tid & 31;
    const int lhalf = lane >> 4;
    const int lmod  = lane & 15;
    const int gbase = blockIdx.x * 128;

    const int   h0   = wave * 16;
    const int   col  = h0 + lmod;
    const float bias = mq_b[col];
    const float* bcol = mq_w + (size_t)col * H_;   // B[k][col] = mq_w[col][k]

    for (int mt = 0; mt < 8; ++mt) {
        const int m0 = mt * 16;
        const float* arow = stacked + (size_t)(gbase + m0 + lmod) * H_ + 2 * lhalf;
        v8f acc = {0.f, 0.f, 0.f, 0.f, 0.f, 0.f, 0.f, 0.f};
        #pragma unroll
        for (int ks = 0; ks < 32; ++ks) {
            const int k = ks * 4;
            v2f a = *(const v2f*)(arow + k);
            v2f b = *(const v2f*)(bcol + k + 2 * lhalf);
            acc = __builtin_amdgcn_wmma_f32_16x16x4_f32(
                false, a, false, b, (short)0, acc, false, false);
        }
        #pragma unroll
        for (int v = 0; v < 8; ++v) {
            const int row = m0 + v + 8 * lhalf;
            const float val = fast_tanh(acc[v] + bias);
            ys[row * 129 + col] = val;
            y[(size_t)(gbase + row) * H_ + col] = val;
        }
    }
    __syncthreads();

    if (tid < 128) {
        float s = mk_b[0];
        for (int k = 0; k < H_; ++k)
            s += ys[tid * 129 + k] * mk_w[k];
        s2[gbase + tid] = s;
    }
}

// ---------------------------------------------------------------------------
// Kernel 3: softmax over 1024 group scores, pool over groups, final proj.
// Single block, 256 threads.
// ---------------------------------------------------------------------------
extern "C" __global__ void __launch_bounds__(256)
k3_final(const float* __restrict__ y, const float* __restrict__ s2,
         const float* __restrict__ out_w, const float* __restrict__ out_b,
         float* __restrict__ out)
{
    __shared__ float red[256];
    __shared__ float ww[1024];
    __shared__ float fin[128];
    const int tid = threadIdx.x;

    float m = -INFINITY;
    for (int i = tid; i < G_; i += 256) m = fmaxf(m, s2[i]);
    red[tid] = m;
    __syncthreads();
    for (int off = 128; off > 0; off >>= 1) {
        if (tid < off) red[tid] = fmaxf(red[tid], red[tid + off]);
        __syncthreads();
    }
    const float mx = red[0];
    __syncthreads();

    float psum = 0.f;
    for (int i = tid; i < G_; i += 256) {
        const float e = __expf(s2[i] - mx);
        ww[i] = e;
        psum += e;
    }
    red[tid] = psum;
    __syncthreads();
    for (int off = 128; off > 0; off >>= 1) {
        if (tid < off) red[tid] += red[tid + off];
        __syncthreads();
    }
    const float Z = red[0];
    __syncthreads();

    const int h    = tid & 127;
    const int half = tid >> 7;
    float a = 0.f;
    for (int g2 = half * 512; g2 < half * 512 + 512; ++g2)
        a += ww[g2] * y[(size_t)g2 * H_ + h];
    red[tid] = a;
    __syncthreads();
    if (tid < 128) fin[tid] = (red[tid] + red[tid + 128]) / Z;
    __syncthreads();

    if (tid < 128) {
        float s = out_b[tid];
        const float* wrow = out_w + (size_t)tid * H_;
        for (int k = 0; k < H_; ++k) s += fin[k] * wrow[k];
        out[tid] = s;
    }
}

// ---------------------------------------------------------------------------
extern "C" void kernel_launch(void* const* d_in, const int* in_sizes, int n_in,
                              void* d_out, int out_size, void* d_ws, size_t ws_size,
                              hipStream_t stream)
{
    const float* feat  = (const float*)d_in[0];
    const float* hq_w  = (const float*)d_in[1];
    const float* hq_b  = (const float*)d_in[2];
    const float* hk_w  = (const float*)d_in[3];
    const float* hk_b  = (const float*)d_in[4];
    const float* mq_w  = (const float*)d_in[5];
    const float* mq_b  = (const float*)d_in[6];
    const float* mk_w  = (const float*)d_in[7];
    const float* mk_b  = (const float*)d_in[8];
    const float* out_w = (const float*)d_in[9];
    const float* out_b = (const float*)d_in[10];
    float* out = (float*)d_out;

    float* ws_stacked = (float*)d_ws;                 // [1024][128]
    float* ws_y       = ws_stacked + G_ * H_;         // [1024][128]
    float* ws_s2      = ws_y + G_ * H_;               // [1024]

    const size_t smem1 = (size_t)(N_ * XS_STRIDE + N_ + 256) * sizeof(float); // ~272KB < 320KB/WGP
    const size_t smem2 = (size_t)(128 * 129) * sizeof(float);                 // ~64.5KB

    k1_group_pool<<<G_, 256, smem1, stream>>>(feat, hq_w, hq_b, hk_w, hk_b, ws_stacked);
    k2_group_mlp<<<8, 256, smem2, stream>>>(ws_stacked, mq_w, mq_b, mk_w, mk_b, ws_y, ws_s2);
    k3_final<<<1, 256, 0, stream>>>(ws_y, ws_s2, out_w, out_b, out);
}